// IouLoss_79688823210006
// MI455X (gfx1250) — compile-verified
//
#include <hip/hip_runtime.h>
#include <hip/hip_bf16.h>
#include <stdint.h>

#define EPSV 1e-8f
#define WAVES_PER_BLOCK 8
#define BLOCK 256

#if defined(__has_builtin)
#if __has_builtin(__builtin_amdgcn_global_load_async_to_lds_b32) && \
    __has_builtin(__builtin_amdgcn_s_wait_asynccnt)
#define HAVE_ASYNC_LDS 1
#endif
#endif

// Intersection area of a convex quad (given in the rect's local frame) with
// the axis-aligned rectangle |x|<=hx, |y|<=hy. Sutherland-Hodgman clipping.
__device__ __forceinline__ float quad_rect_inter_area(const float* rx,
                                                      const float* ry,
                                                      float hx, float hy) {
    float px[8], py[8], qx[8], qy[8];
    int n = 4;
#pragma unroll
    for (int k = 0; k < 4; ++k) { px[k] = rx[k]; py[k] = ry[k]; }

#pragma unroll
    for (int e = 0; e < 4; ++e) {
        int m = 0;
        for (int k = 0; k < n; ++k) {
            int k2 = (k + 1 == n) ? 0 : k + 1;
            float cx = px[k],  cy = py[k];
            float nx = px[k2], ny = py[k2];
            float dc, dn;
            if (e == 0)      { dc = cx + hx; dn = nx + hx; }   // x >= -hx
            else if (e == 1) { dc = hx - cx; dn = hx - nx; }   // x <=  hx
            else if (e == 2) { dc = cy + hy; dn = ny + hy; }   // y >= -hy
            else             { dc = hy - cy; dn = hy - ny; }   // y <=  hy
            bool cin = (dc >= 0.0f), nin = (dn >= 0.0f);
            if (cin) { qx[m] = cx; qy[m] = cy; ++m; }
            if (cin != nin) {
                float t = dc / (dc - dn);
                qx[m] = cx + t * (nx - cx);
                qy[m] = cy + t * (ny - cy);
                ++m;
            }
        }
        n = m;
        if (n == 0) return 0.0f;
        for (int k = 0; k < n; ++k) { px[k] = qx[k]; py[k] = qy[k]; }
    }

    float area = 0.0f;
    for (int k = 0; k < n; ++k) {
        int k2 = (k + 1 == n) ? 0 : k + 1;
        area += px[k] * py[k2] - px[k2] * py[k];
    }
    return 0.5f * fabsf(area);
}

// One wave (32 lanes) per predicted box; lanes stride over the M GT boxes.
extern "C" __global__ __launch_bounds__(BLOCK)
void iou_loss_main(const float* __restrict__ iou_pred,
                   const float* __restrict__ box_pred,
                   const float* __restrict__ box_gt,
                   float* __restrict__ partials,
                   int N, int M) {
    extern __shared__ float smem[];        // [M*7] raw gt | [M*8] derived gt
    float* raw = smem;
    float* gt  = smem + M * 7;
    __shared__ float wsum[WAVES_PER_BLOCK];

    const int tid = threadIdx.x;

    // ---- Stage raw GT boxes into LDS (async-to-LDS path on gfx1250) ----
#if HAVE_ASYNC_LDS
    {
        const int total = M * 7;
        for (int idx = tid; idx < total; idx += BLOCK) {
            __builtin_amdgcn_global_load_async_to_lds_b32(
                (__attribute__((address_space(1))) int*)(box_gt + idx),
                (__attribute__((address_space(3))) int*)(raw + idx),
                0, 0);
        }
        __builtin_amdgcn_s_wait_asynccnt(0);
    }
#else
    for (int idx = tid; idx < M * 7; idx += BLOCK) raw[idx] = box_gt[idx];
#endif
    __syncthreads();

    // ---- Derive per-GT data: center, cos, sin, hx, hy, area ----
    for (int j = tid; j < M; j += BLOCK) {
        const float* b = raw + j * 7;
        float bx = b[0], by = b[1], dx = b[3], dy = b[4], yaw = b[6];
        float c = __cosf(yaw), s = __sinf(yaw);
        float* g = gt + j * 8;
        g[0] = bx; g[1] = by; g[2] = c; g[3] = s;
        g[4] = dx * 0.5f; g[5] = dy * 0.5f; g[6] = dx * dy; g[7] = 0.0f;
    }
    __syncthreads();

    const int wave = tid >> 5;
    const int lane = tid & 31;
    const int i = blockIdx.x * WAVES_PER_BLOCK + wave;

    float contrib = 0.0f;
    if (i < N) {
        const float* bp = box_pred + i * 7;
        float ax = bp[0], ay = bp[1], adx = bp[3], ady = bp[4], ayaw = bp[6];
        float ac = __cosf(ayaw), as = __sinf(ayaw);
        float ahx = adx * 0.5f, ahy = ady * 0.5f;
        float areaA = adx * ady;

        // world-space corners of pred box (CCW, matches reference ordering)
        const float lxs[4] = { ahx, -ahx, -ahx,  ahx };
        const float lys[4] = { ahy,  ahy, -ahy, -ahy };
        float wx[4], wy[4];
#pragma unroll
        for (int k = 0; k < 4; ++k) {
            wx[k] = ax + ac * lxs[k] - as * lys[k];
            wy[k] = ay + as * lxs[k] + ac * lys[k];
        }

        float best = 0.0f;
        for (int j = lane; j < M; j += 32) {
            const float* g = gt + j * 8;
            float bx = g[0], by = g[1], bc = g[2], bs = g[3];
            float bhx = g[4], bhy = g[5], areaB = g[6];
            // pred corners in GT local frame
            float rx[4], ry[4];
#pragma unroll
            for (int k = 0; k < 4; ++k) {
                float dxp = wx[k] - bx, dyp = wy[k] - by;
                rx[k] =  bc * dxp + bs * dyp;
                ry[k] = -bs * dxp + bc * dyp;
            }
            float inter = quad_rect_inter_area(rx, ry, bhx, bhy);
            float uni = fmaxf(areaA + areaB - inter, EPSV);
            best = fmaxf(best, inter / uni);
        }

        // wave32 max reduction
#pragma unroll
        for (int off = 16; off >= 1; off >>= 1)
            best = fmaxf(best, __shfl_xor(best, off, 32));

        if (lane == 0) {
            float target = 2.0f * best - 1.0f;
            contrib = fabsf(iou_pred[i] - target);
        }
    }

    if (lane == 0) wsum[wave] = contrib;
    __syncthreads();
    if (tid == 0) {
        float s = 0.0f;
#pragma unroll
        for (int w = 0; w < WAVES_PER_BLOCK; ++w) s += wsum[w];
        partials[blockIdx.x] = s;
    }
}

extern "C" __global__ __launch_bounds__(256)
void iou_loss_finalize(const float* __restrict__ partials, int nparts,
                       const int* __restrict__ ntp, float* __restrict__ out) {
    __shared__ float sh[256];
    float s = 0.0f;
    for (int idx = threadIdx.x; idx < nparts; idx += 256) s += partials[idx];
    sh[threadIdx.x] = s;
    __syncthreads();
    for (int stride = 128; stride > 0; stride >>= 1) {
        if (threadIdx.x < stride) sh[threadIdx.x] += sh[threadIdx.x + stride];
        __syncthreads();
    }
    if (threadIdx.x == 0) out[0] = sh[0] / ((float)ntp[0] + 1e-4f);
}

extern "C" void kernel_launch(void* const* d_in, const int* in_sizes, int n_in,
                              void* d_out, int out_size, void* d_ws, size_t ws_size,
                              hipStream_t stream) {
    const float* iou_pred = (const float*)d_in[0];
    const float* box_pred = (const float*)d_in[1];
    const float* box_gt   = (const float*)d_in[2];
    const int*   ntp      = (const int*)d_in[3];

    const int N = in_sizes[1] / 7;
    const int M = in_sizes[2] / 7;

    float* partials = (float*)d_ws;
    const int nblocks = (N + WAVES_PER_BLOCK - 1) / WAVES_PER_BLOCK;
    const size_t shmem = (size_t)M * (7 + 8) * sizeof(float);

    hipLaunchKernelGGL(iou_loss_main, dim3(nblocks), dim3(BLOCK), shmem, stream,
                       iou_pred, box_pred, box_gt, partials, N, M);
    hipLaunchKernelGGL(iou_loss_finalize, dim3(1), dim3(256), 0, stream,
                       partials, nblocks, ntp, (float*)d_out);
}